// DeepSSM_42898133353244
// MI455X (gfx1250) — compile-verified
//
#include <hip/hip_runtime.h>
#include <math.h>

// ---------------- model dims ----------------
#define D_MODEL 256
#define SEQ     2048
#define BATCH   8
#define NLAYERS 8
#define ROWS    (BATCH*SEQ)      // 16384 = BATCH*SEQ
#define NCH     16               // scan chunks per sequence
#define CL      (SEQ/NCH)        // 128 steps per chunk

typedef float v2f __attribute__((ext_vector_type(2)));
typedef float v8f __attribute__((ext_vector_type(8)));

__device__ __forceinline__ float gelu_exact(float x) {
    // jax.nn.gelu(approximate=False) = 0.5*x*(1+erf(x/sqrt(2)))
    return 0.5f * x * (1.0f + erff(x * 0.70710678118654752440f));
}

// ---------------- kernel 0: input projection (INPUT_DIM==1) ----------------
__global__ void k_inproj(const float* __restrict__ x, const float* __restrict__ W,
                         const float* __restrict__ b, float* __restrict__ H) {
    int row = blockIdx.x;          // b*SEQ + l
    int d   = threadIdx.x;
    H[(size_t)row * D_MODEL + d] = x[row] * W[d] + b[d];
}

// ---------------- scan pass 1: per-chunk sums of A^s * x[s] ----------------
__global__ void k_scan_part(const float* __restrict__ H, const float* __restrict__ Ap,
                            float* __restrict__ S) {
    int b = blockIdx.x / NCH, c = blockIdx.x % NCH;
    int d = threadIdx.x;
    float A   = 1.0f / (1.0f + expf(-Ap[d]));   // sigmoid
    float l2A = log2f(A);
    int   s0  = c * CL;
    float w   = exp2f((float)s0 * l2A);          // A^s0
    float sum = 0.0f;
    const float* hp = H + ((size_t)b * SEQ + s0) * D_MODEL + d;
    #pragma unroll 4
    for (int j = 0; j < CL; ++j) {
        sum += w * hp[j * D_MODEL];
        w *= A;
    }
    S[(b * NCH + c) * D_MODEL + d] = sum;
}

// ---------------- scan pass 2: exclusive prefix over chunks ----------------
__global__ void k_scan_prefix(const float* __restrict__ S, float* __restrict__ O) {
    int b = blockIdx.x, d = threadIdx.x;
    float acc = 0.0f;
    for (int c = 0; c < NCH; ++c) {
        int idx = (b * NCH + c) * D_MODEL + d;
        O[idx] = acc;
        acc += S[idx];
    }
}

// ---------------- scan pass 3: out = GELU(CB*A^(L-1-t)*P[t] + D*x) ----------------
__global__ void k_scan_apply(const float* __restrict__ H, const float* __restrict__ O,
                             const float* __restrict__ Ap, const float* __restrict__ Bp,
                             const float* __restrict__ Cp, const float* __restrict__ Dp,
                             float* __restrict__ G) {
    int b = blockIdx.x / NCH, c = blockIdx.x % NCH;
    int d = threadIdx.x;
    float A   = 1.0f / (1.0f + expf(-Ap[d]));
    float l2A = log2f(A);
    float CB  = Cp[d] * Bp[d];
    float Dv  = Dp[d];
    int   s0  = c * CL;
    float w   = exp2f((float)s0 * l2A);
    float P   = O[(b * NCH + c) * D_MODEL + d];
    const float* hp = H + ((size_t)b * SEQ + s0) * D_MODEL + d;
    float*       gp = G + ((size_t)b * SEQ + s0) * D_MODEL + d;
    for (int j = 0; j < CL; ++j) {
        int s = s0 + j;
        float xv = hp[j * D_MODEL];
        P += w * xv;
        float invw = exp2f((float)(SEQ - 1 - s) * l2A);  // A^(L-1-t)
        float o = CB * invw * P + Dv * xv;
        gp[j * D_MODEL] = gelu_exact(o);
        w *= A;
    }
}

// ---------------- GEMM: Y = G @ pW^T + pb + H (residual) via f32 WMMA ----------------
// M=16384, N=256, K=256. Block = 8 waves; block tile 128Mx64N; wave tile 16Mx64N.
// A in LDS as floats [m][k] (padded 68); B in LDS as K-pair float2s: Bs2[k/2][n]
// with row stride 81 float2 (648B) -> frag reads are single b64 DS loads, halves of
// the wave (rows kp / kp+1) hit disjoint bank sets, staging writes are conflict-free.
__global__ __launch_bounds__(256)
void k_gemm_wmma(const float* __restrict__ G, const float* __restrict__ pW,
                 const float* __restrict__ pb, const float* __restrict__ H,
                 float* __restrict__ Y) {
    __shared__ float As[128][68];
    __shared__ v2f   Bs2[32][81];

    int tid  = threadIdx.x;
    int wv   = tid >> 5;
    int lane = tid & 31;
    int half = lane >> 4;           // 0: lanes 0-15, 1: lanes 16-31
    int lm   = lane & 15;
    int m0   = blockIdx.x * 128;
    int n0   = blockIdx.y * 64;

    v8f acc0 = {}, acc1 = {}, acc2 = {}, acc3 = {};

    for (int kb = 0; kb < D_MODEL; kb += 64) {
        // stage A: K-pairs as float2 -> coalesced 256B/wave global b64 loads
        for (int idx = tid; idx < 128 * 32; idx += 256) {
            int p = idx & 31, r = idx >> 5;
            *(v2f*)&As[r][2 * p] =
                *(const v2f*)&G[(size_t)(m0 + r) * D_MODEL + kb + 2 * p];
        }
        // stage B: Bs2[p][n] = {pW[(n0+n)*256+kb+2p], pW[...+2p+1]}
        for (int idx = tid; idx < 32 * 64; idx += 256) {
            int p = idx & 31, n = idx >> 5;
            Bs2[p][n] = *(const v2f*)&pW[(size_t)(n0 + n) * D_MODEL + kb + 2 * p];
        }
        __syncthreads();

        // prefetch next K-slice while this stage computes (global_prefetch_b8)
        if (kb + 64 < D_MODEL) {
            __builtin_prefetch(&G[(size_t)(m0 + (tid >> 1)) * D_MODEL + kb + 64 + (tid & 1) * 32], 0, 1);
            __builtin_prefetch(&pW[(size_t)(n0 + (tid >> 2)) * D_MODEL + kb + 64 + (tid & 3) * 16], 0, 1);
        }

        for (int kk = 0; kk < 64; kk += 4) {
            int ka = kk + 2 * half;             // ISA A-layout: K = vgpr + 2*half
            int kp = (kk >> 1) + half;          // K-pair row for this half-wave
            v2f a;
            a.x = As[wv * 16 + lm][ka];
            a.y = As[wv * 16 + lm][ka + 1];
            v2f b0 = Bs2[kp][lm];
            v2f b1 = Bs2[kp][16 + lm];
            v2f b2 = Bs2[kp][32 + lm];
            v2f b3 = Bs2[kp][48 + lm];
            acc0 = __builtin_amdgcn_wmma_f32_16x16x4_f32(false, a, false, b0, (short)0, acc0, false, false);
            acc1 = __builtin_amdgcn_wmma_f32_16x16x4_f32(false, a, false, b1, (short)0, acc1, false, false);
            acc2 = __builtin_amdgcn_wmma_f32_16x16x4_f32(false, a, false, b2, (short)0, acc2, false, false);
            acc3 = __builtin_amdgcn_wmma_f32_16x16x4_f32(false, a, false, b3, (short)0, acc3, false, false);
        }
        __syncthreads();
    }

    // epilogue: D layout -> M = vgpr + 8*half, N = lane%16 (per 16-wide n-subtile)
    int mrow0 = m0 + wv * 16 + 8 * half;
    #pragma unroll
    for (int r = 0; r < 8; ++r) {
        int m = mrow0 + r;
        size_t base = (size_t)m * D_MODEL;
        int n;
        n = n0 + lm;      Y[base + n] = acc0[r] + pb[n] + H[base + n];
        n = n0 + 16 + lm; Y[base + n] = acc1[r] + pb[n] + H[base + n];
        n = n0 + 32 + lm; Y[base + n] = acc2[r] + pb[n] + H[base + n];
        n = n0 + 48 + lm; Y[base + n] = acc3[r] + pb[n] + H[base + n];
    }
}

// ---------------- LayerNorm over d=256: one wave per row ----------------
__global__ void k_ln(const float* __restrict__ Y, const float* __restrict__ g,
                     const float* __restrict__ bta, float* __restrict__ H) {
    int lane = threadIdx.x & 31;
    int row  = blockIdx.x * 8 + (threadIdx.x >> 5);
    const float* yp = Y + (size_t)row * D_MODEL;
    float v[8];
    float s = 0.0f;
    #pragma unroll
    for (int j = 0; j < 8; ++j) { v[j] = yp[lane + 32 * j]; s += v[j]; }
    #pragma unroll
    for (int m = 16; m >= 1; m >>= 1) s += __shfl_xor(s, m, 32);
    float mu = s * (1.0f / 256.0f);
    float q = 0.0f;
    #pragma unroll
    for (int j = 0; j < 8; ++j) { float t = v[j] - mu; q += t * t; }
    #pragma unroll
    for (int m = 16; m >= 1; m >>= 1) q += __shfl_xor(q, m, 32);
    float rstd = rsqrtf(q * (1.0f / 256.0f) + 1e-5f);
    float* hp = H + (size_t)row * D_MODEL;
    #pragma unroll
    for (int j = 0; j < 8; ++j) {
        int dd = lane + 32 * j;
        hp[dd] = (v[j] - mu) * rstd * g[dd] + bta[dd];
    }
}

// ---------------- exit head: mean-pool over L, then [3,256] matvec ----------------
__global__ void k_head(const float* __restrict__ H, const float* __restrict__ hW,
                       const float* __restrict__ hb, float* __restrict__ out, int head) {
    __shared__ float red[256];
    int b = blockIdx.x, d = threadIdx.x;
    float s = 0.0f;
    const float* hp = H + (size_t)b * SEQ * D_MODEL + d;
    for (int l = 0; l < SEQ; ++l) s += hp[(size_t)l * D_MODEL];
    float pooled = s * (1.0f / (float)SEQ);
    for (int c = 0; c < 3; ++c) {
        __syncthreads();
        red[d] = pooled * hW[c * D_MODEL + d];
        __syncthreads();
        for (int off = 128; off > 0; off >>= 1) {
            if (d < off) red[d] += red[d + off];
            __syncthreads();
        }
        if (d == 0) out[head * (BATCH * 3) + b * 3 + c] = red[0] + hb[c];
    }
}

// ---------------- launcher ----------------
extern "C" void kernel_launch(void* const* d_in, const int* in_sizes, int n_in,
                              void* d_out, int out_size, void* d_ws, size_t ws_size,
                              hipStream_t stream) {
    const float* x   = (const float*)d_in[0];
    const float* inW = (const float*)d_in[1];
    const float* inb = (const float*)d_in[2];
    const float* Ap  = (const float*)d_in[3];
    const float* Bp  = (const float*)d_in[4];
    const float* Cp  = (const float*)d_in[5];
    const float* Dp  = (const float*)d_in[6];
    const float* pW  = (const float*)d_in[7];
    const float* pb  = (const float*)d_in[8];
    const float* lg  = (const float*)d_in[9];
    const float* lb  = (const float*)d_in[10];
    const float* hW  = (const float*)d_in[11];
    const float* hb  = (const float*)d_in[12];
    float* out = (float*)d_out;

    const size_t ACT = (size_t)ROWS * D_MODEL;   // 4,194,304 floats
    float* H = (float*)d_ws;
    float* G = H + ACT;
    float* Y = G + ACT;
    float* S = Y + ACT;
    float* O = S + (size_t)BATCH * NCH * D_MODEL;

    k_inproj<<<ROWS, D_MODEL, 0, stream>>>(x, inW, inb, H);

    int head = 0;
    for (int i = 0; i < NLAYERS; ++i) {
        const float* Ai = Ap + i * D_MODEL;
        k_scan_part  <<<BATCH * NCH, D_MODEL, 0, stream>>>(H, Ai, S);
        k_scan_prefix<<<BATCH,       D_MODEL, 0, stream>>>(S, O);
        k_scan_apply <<<BATCH * NCH, D_MODEL, 0, stream>>>(H, O, Ai, Bp + i * D_MODEL,
                                                           Cp + i * D_MODEL, Dp + i * D_MODEL, G);
        dim3 gdim(ROWS / 128, D_MODEL / 64);
        k_gemm_wmma<<<gdim, 256, 0, stream>>>(G, pW + (size_t)i * D_MODEL * D_MODEL,
                                              pb + i * D_MODEL, H, Y);
        k_ln<<<ROWS / 8, 256, 0, stream>>>(Y, lg + i * D_MODEL, lb + i * D_MODEL, H);
        if (i == 1 || i == 3 || i == 5 || i == 7) {
            k_head<<<BATCH, D_MODEL, 0, stream>>>(H, hW + head * 3 * D_MODEL, hb + head * 3, out, head);
            ++head;
        }
    }
}